// GATModel_75213467288112
// MI455X (gfx1250) — compile-verified
//
#include <hip/hip_runtime.h>
#include <math.h>

// ---------------------------------------------------------------------------
// GAT forward for MI455X (gfx1250).  Dense projections use
// V_WMMA_F32_16X16X4_F32 (fp32 A/B, fp32 accumulate) with an LDS-staged B
// tile and branch-free clamped loads.  Edge softmax + segment-sum use
// L2-resident float atomics (out1/xp1 each ~102MB < 192MB L2).
// ---------------------------------------------------------------------------

typedef __attribute__((ext_vector_type(2))) float v2f;
typedef __attribute__((ext_vector_type(8))) float v8f;

#define NEG_SLOPE 0.2f
#define F_IN   166
#define HID    64
#define HEADS  8
#define N_CLS  2

// ---------------- utility: float atomic max via integer punning -------------
__device__ __forceinline__ void atomic_max_f32(float* addr, float val) {
  if (val >= 0.0f)
    atomicMax((int*)addr, __float_as_int(val));
  else
    atomicMin((unsigned int*)addr, __float_as_uint(val));
}

__device__ __forceinline__ void edge_nodes(const long long* __restrict__ ei,
                                           int E, int e, int& s, int& d) {
  if (e < E) {
    s = (int)ei[e];
    d = (int)ei[(size_t)E + e];
  } else {            // self loop appended after the E real edges
    s = d = e - E;
  }
}

// ---------------- fill ------------------------------------------------------
__global__ void fill_f32(float* __restrict__ p, long long n, float v) {
  long long i = (long long)blockIdx.x * blockDim.x + threadIdx.x;
  if (i < n) p[i] = v;
}

// ---------------- WMMA fp32 GEMM:  C[M,N] = act(A[M,K] @ B[K,N] + bias) -----
// block = 256 threads = 8 waves; wave -> 16(M) x 64(N) tile; block -> 128x64.
// B is staged through LDS once per block (transposed [n][k] so each lane pulls
// its (k,k+1) pair with one aligned ds_load_b64).  All loads in the main K
// loop are unguarded: M/N out-of-range lanes use clamped addresses and their
// results are simply never stored (WMMA rows<->A lanes, cols<->B lanes).
// act: 0 = none, 1 = relu
__global__ void __launch_bounds__(256)
gemm_wmma_f32(const float* __restrict__ A,
              const float* __restrict__ B,
              float* __restrict__ C,
              int M, int K, int N,
              const float* __restrict__ bias, int act) {
  __shared__ float Bs[64][4];              // [n within tile][k within step]

  const int tid   = threadIdx.x;
  const int wave  = tid >> 5;
  const int lane  = tid & 31;
  const int lhalf = lane & 15;             // position within half-wave
  const int khalf = lane >> 4;             // 0 -> K+0/1 , 1 -> K+2/3

  const int m0 = blockIdx.x * 128 + wave * 16;
  const int n0 = blockIdx.y * 64;

  // cooperative B staging coords: thread i loads B[k0 + (i>>6)][n0 + (i&63)]
  const int bk   = tid >> 6;                         // 0..3
  const int bn   = tid & 63;                         // 0..63
  const int bn_c = min(n0 + bn, N - 1);              // clamped column
  const bool bn_ok = (n0 + bn) < N;

  const int am = m0 + lhalf;
  const float* __restrict__ Arow = A + (size_t)min(am, M - 1) * K;

  v8f acc0 = {}, acc1 = {}, acc2 = {}, acc3 = {};

  const int kmain = K & ~3;
  int k0 = 0;
  for (; k0 < kmain; k0 += 4) {
    float bv = B[(size_t)(k0 + bk) * N + bn_c];
    if (!bn_ok) bv = 0.0f;
    __syncthreads();                       // previous iteration done reading
    Bs[bn][bk] = bv;
    __syncthreads();

    const int ak = k0 + khalf * 2;
    const v2f a  = *(const v2f*)(Arow + ak);
    const v2f b0 = *(const v2f*)(&Bs[ 0 + lhalf][khalf * 2]);
    const v2f b1 = *(const v2f*)(&Bs[16 + lhalf][khalf * 2]);
    const v2f b2 = *(const v2f*)(&Bs[32 + lhalf][khalf * 2]);
    const v2f b3 = *(const v2f*)(&Bs[48 + lhalf][khalf * 2]);
    acc0 = __builtin_amdgcn_wmma_f32_16x16x4_f32(false, a, false, b0, (short)0, acc0, false, false);
    acc1 = __builtin_amdgcn_wmma_f32_16x16x4_f32(false, a, false, b1, (short)0, acc1, false, false);
    acc2 = __builtin_amdgcn_wmma_f32_16x16x4_f32(false, a, false, b2, (short)0, acc2, false, false);
    acc3 = __builtin_amdgcn_wmma_f32_16x16x4_f32(false, a, false, b3, (short)0, acc3, false, false);
  }

  if (k0 < K) {                            // K tail (K not multiple of 4)
    const int kk = k0 + bk;
    float bv = B[(size_t)min(kk, K - 1) * N + bn_c];
    if (!(kk < K && bn_ok)) bv = 0.0f;
    __syncthreads();
    Bs[bn][bk] = bv;
    __syncthreads();

    const int ak = k0 + khalf * 2;
    float ax = Arow[min(ak,     K - 1)];
    float ay = Arow[min(ak + 1, K - 1)];
    if (ak     >= K) ax = 0.0f;
    if (ak + 1 >= K) ay = 0.0f;
    v2f a; a.x = ax; a.y = ay;
    const v2f b0 = *(const v2f*)(&Bs[ 0 + lhalf][khalf * 2]);
    const v2f b1 = *(const v2f*)(&Bs[16 + lhalf][khalf * 2]);
    const v2f b2 = *(const v2f*)(&Bs[32 + lhalf][khalf * 2]);
    const v2f b3 = *(const v2f*)(&Bs[48 + lhalf][khalf * 2]);
    acc0 = __builtin_amdgcn_wmma_f32_16x16x4_f32(false, a, false, b0, (short)0, acc0, false, false);
    acc1 = __builtin_amdgcn_wmma_f32_16x16x4_f32(false, a, false, b1, (short)0, acc1, false, false);
    acc2 = __builtin_amdgcn_wmma_f32_16x16x4_f32(false, a, false, b2, (short)0, acc2, false, false);
    acc3 = __builtin_amdgcn_wmma_f32_16x16x4_f32(false, a, false, b3, (short)0, acc3, false, false);
  }

  // C/D layout: VGPR r -> lanes 0-15 row m0+r, lanes 16-31 row m0+8+r
  const int rbase = m0 + khalf * 8;
  v8f accs[4] = {acc0, acc1, acc2, acc3};
#pragma unroll
  for (int t = 0; t < 4; ++t) {
    const int col = n0 + t * 16 + lhalf;
    if (col < N) {
      const float bb = bias ? bias[col] : 0.0f;
#pragma unroll
      for (int r = 0; r < 8; ++r) {
        const int row = rbase + r;
        if (row < M) {
          float v = accs[t][r] + bb;
          if (act == 1) v = fmaxf(v, 0.0f);
          C[(size_t)row * N + col] = v;
        }
      }
    }
  }
}

// ---------------- attention scores:  al_s/al_d[n*H+h] = <xp[n,h,:], a_*[h,:]>
__global__ void att_scores(const float* __restrict__ xp,
                           const float* __restrict__ a_src,
                           const float* __restrict__ a_dst,
                           float* __restrict__ al_s,
                           float* __restrict__ al_d,
                           int N, int H, int C) {
  const int i = blockIdx.x * blockDim.x + threadIdx.x;   // n*H + h
  if (i >= N * H) return;
  const int h = i % H;
  const float* __restrict__ xr = xp + (size_t)i * C;     // (n*H+h)*C
  const float* __restrict__ as = a_src + h * C;
  const float* __restrict__ ad = a_dst + h * C;
  float s = 0.0f, d = 0.0f;
  for (int c = 0; c < C; ++c) {
    const float v = xr[c];
    s += v * as[c];
    d += v * ad[c];
  }
  al_s[i] = s;
  al_d[i] = d;
}

// ---------------- edge pass A: segment max of leaky-relu'ed logits ----------
__global__ void edge_max(const long long* __restrict__ ei,
                         const float* __restrict__ al_s,
                         const float* __restrict__ al_d,
                         float* __restrict__ m,
                         int E, int Nn, int H) {
  const int e = blockIdx.x * blockDim.x + threadIdx.x;
  if (e >= E + Nn) return;
  int s, d;
  edge_nodes(ei, E, e, s, d);
  for (int h = 0; h < H; ++h) {
    float v = al_s[(size_t)s * H + h] + al_d[(size_t)d * H + h];
    v = (v > 0.0f) ? v : NEG_SLOPE * v;
    atomic_max_f32(&m[(size_t)d * H + h], v);
  }
}

// ---------------- edge pass B: ex = exp(e - m[d]);  denom[d] += ex ----------
__global__ void edge_expsum(const long long* __restrict__ ei,
                            const float* __restrict__ al_s,
                            const float* __restrict__ al_d,
                            const float* __restrict__ m,
                            float* __restrict__ ex,
                            float* __restrict__ denom,
                            int E, int Nn, int H) {
  const int e = blockIdx.x * blockDim.x + threadIdx.x;
  if (e >= E + Nn) return;
  int s, d;
  edge_nodes(ei, E, e, s, d);
  for (int h = 0; h < H; ++h) {
    float v = al_s[(size_t)s * H + h] + al_d[(size_t)d * H + h];
    v = (v > 0.0f) ? v : NEG_SLOPE * v;
    const float t = expf(v - m[(size_t)d * H + h]);
    ex[(size_t)e * H + h] = t;
    atomicAdd(&denom[(size_t)d * H + h], t);
  }
}

// ---------------- edge pass C: out[d,h,c] += (ex/denom[d,h]) * xp[s,h,c] ----
__global__ void edge_aggregate(const long long* __restrict__ ei,
                               const float* __restrict__ xp,
                               const float* __restrict__ ex,
                               const float* __restrict__ denom,
                               float* __restrict__ out,
                               int E, int Nn, int H, int C) {
  const long long gid   = (long long)blockIdx.x * blockDim.x + threadIdx.x;
  const long long total = (long long)(E + Nn) * H * C;
  if (gid >= total) return;
  const int c   = (int)(gid % C);
  const long long r1 = gid / C;
  const int h   = (int)(r1 % H);
  const int e   = (int)(r1 / H);
  int s, d;
  edge_nodes(ei, E, e, s, d);
  const float alpha = ex[(size_t)e * H + h] / denom[(size_t)d * H + h];
  const float v = alpha * xp[((size_t)s * H + h) * C + c];
  atomicAdd(&out[((size_t)d * H + h) * C + c], v);
}

// ---------------- elementwise bias + activation (in place) ------------------
// act: 0 = none, 2 = elu
__global__ void bias_act(float* __restrict__ x, const float* __restrict__ b,
                         long long total, int N, int act) {
  const long long i = (long long)blockIdx.x * blockDim.x + threadIdx.x;
  if (i >= total) return;
  float v = x[i] + b[(int)(i % N)];
  if (act == 2) v = (v > 0.0f) ? v : (expf(v) - 1.0f);   // elu(alpha=1)
  x[i] = v;
}

// ---------------------------------------------------------------------------
extern "C" void kernel_launch(void* const* d_in, const int* in_sizes, int n_in,
                              void* d_out, int out_size, void* d_ws, size_t ws_size,
                              hipStream_t stream) {
  const float*     x      = (const float*)d_in[0];
  const long long* ei     = (const long long*)d_in[1];   // int64 [2,E]
  const float*     W1     = (const float*)d_in[2];
  const float*     a_src1 = (const float*)d_in[3];
  const float*     a_dst1 = (const float*)d_in[4];
  const float*     b1     = (const float*)d_in[5];
  const float*     W2     = (const float*)d_in[6];
  const float*     a_src2 = (const float*)d_in[7];
  const float*     a_dst2 = (const float*)d_in[8];
  const float*     b2     = (const float*)d_in[9];
  const float*     Wc1    = (const float*)d_in[10];
  const float*     bc1    = (const float*)d_in[11];
  const float*     Wc2    = (const float*)d_in[12];
  const float*     bc2    = (const float*)d_in[13];
  float*           out    = (float*)d_out;

  const int N   = in_sizes[0] / F_IN;     // 50000
  const int E   = in_sizes[1] / 2;        // 800000
  const int Et  = E + N;                  // edges + self loops
  const int HC  = HEADS * HID;            // 512

  // ---- workspace layout (floats), with aliasing across phases ----
  float* ws   = (float*)d_ws;
  float* xp1  = ws;                          // [N,512]  (reused later)
  float* agg1 = ws + (size_t)N * HC;         // [N,512]  -> h1 in place
  float* arena = agg1 + (size_t)N * HC;

  // layer-1 arena
  float* al_s1 = arena;                              // [N,8]
  float* al_d1 = al_s1 + (size_t)N * HEADS;          // [N,8]
  float* m1    = al_d1 + (size_t)N * HEADS;          // [N,8]
  float* den1  = m1    + (size_t)N * HEADS;          // [N,8]
  float* ex1   = den1  + (size_t)N * HEADS;          // [Et,8]

  // layer-2 aliases (layer-1 arena dead after pass C; xp1 dead after pass C)
  float* xp2   = xp1;                                // [N,64]
  float* agg2  = xp1 + (size_t)N * HID;              // [N,64] -> h2 in place
  float* hc    = agg2 + (size_t)N * HID;             // [N,64]
  float* al_s2 = arena;                              // [N,1]
  float* al_d2 = al_s2 + N;
  float* m2    = al_d2 + N;
  float* den2  = m2 + N;
  float* ex2   = den2 + N;                           // [Et,1]

  const int TB = 256;
  const float NEG_INF = -__builtin_inff();

  // ================= layer 1: GAT 166 -> 64 x 8 heads, concat ==============
  {
    long long n8 = (long long)N * HEADS, nA = (long long)N * HC;
    fill_f32<<<dim3((unsigned)((n8 + TB - 1) / TB)), TB, 0, stream>>>(m1, n8, NEG_INF);
    fill_f32<<<dim3((unsigned)((n8 + TB - 1) / TB)), TB, 0, stream>>>(den1, n8, 0.0f);
    fill_f32<<<dim3((unsigned)((nA + TB - 1) / TB)), TB, 0, stream>>>(agg1, nA, 0.0f);
  }
  gemm_wmma_f32<<<dim3((N + 127) / 128, HC / 64), TB, 0, stream>>>(
      x, W1, xp1, N, F_IN, HC, nullptr, 0);
  att_scores<<<(N * HEADS + TB - 1) / TB, TB, 0, stream>>>(
      xp1, a_src1, a_dst1, al_s1, al_d1, N, HEADS, HID);
  edge_max<<<(Et + TB - 1) / TB, TB, 0, stream>>>(ei, al_s1, al_d1, m1, E, N, HEADS);
  edge_expsum<<<(Et + TB - 1) / TB, TB, 0, stream>>>(
      ei, al_s1, al_d1, m1, ex1, den1, E, N, HEADS);
  {
    long long total = (long long)Et * HEADS * HID;
    edge_aggregate<<<dim3((unsigned)((total + TB - 1) / TB)), TB, 0, stream>>>(
        ei, xp1, ex1, den1, agg1, E, N, HEADS, HID);
  }
  {  // h1 = elu(agg1 + b1)   (in place)
    long long total = (long long)N * HC;
    bias_act<<<dim3((unsigned)((total + TB - 1) / TB)), TB, 0, stream>>>(
        agg1, b1, total, HC, 2);
  }
  float* h1 = agg1;

  // ================= layer 2: GAT 512 -> 64, 1 head, mean ==================
  {
    long long n1 = N, nA = (long long)N * HID;
    fill_f32<<<dim3((unsigned)((n1 + TB - 1) / TB)), TB, 0, stream>>>(m2, n1, NEG_INF);
    fill_f32<<<dim3((unsigned)((n1 + TB - 1) / TB)), TB, 0, stream>>>(den2, n1, 0.0f);
    fill_f32<<<dim3((unsigned)((nA + TB - 1) / TB)), TB, 0, stream>>>(agg2, nA, 0.0f);
  }
  gemm_wmma_f32<<<dim3((N + 127) / 128, HID / 64), TB, 0, stream>>>(
      h1, W2, xp2, N, HC, HID, nullptr, 0);
  att_scores<<<(N + TB - 1) / TB, TB, 0, stream>>>(
      xp2, a_src2, a_dst2, al_s2, al_d2, N, 1, HID);
  edge_max<<<(Et + TB - 1) / TB, TB, 0, stream>>>(ei, al_s2, al_d2, m2, E, N, 1);
  edge_expsum<<<(Et + TB - 1) / TB, TB, 0, stream>>>(
      ei, al_s2, al_d2, m2, ex2, den2, E, N, 1);
  {
    long long total = (long long)Et * HID;
    edge_aggregate<<<dim3((unsigned)((total + TB - 1) / TB)), TB, 0, stream>>>(
        ei, xp2, ex2, den2, agg2, E, N, 1, HID);
  }
  {  // h2 = agg2 + b2  (mean over 1 head == identity)
    long long total = (long long)N * HID;
    bias_act<<<dim3((unsigned)((total + TB - 1) / TB)), TB, 0, stream>>>(
        agg2, b2, total, HID, 0);
  }
  float* h2 = agg2;

  // ================= classifier MLP 64 -> 64 -> 2 ==========================
  gemm_wmma_f32<<<dim3((N + 127) / 128, 1), TB, 0, stream>>>(
      h2, Wc1, hc, N, HID, HID, bc1, 1);                 // relu fused
  gemm_wmma_f32<<<dim3((N + 127) / 128, 1), TB, 0, stream>>>(
      hc, Wc2, out, N, HID, N_CLS, bc2, 0);
}